// DhnLoss_32152125177948
// MI455X (gfx1250) — compile-verified
//
#include <hip/hip_runtime.h>
#include <stdint.h>

#define NM 64
#define NQ 1024
#define NT 1024
#define DELTA 0.1f
#define SCALE 100.0f

// ---- column-pass tiling ----
#define TTILE   128            // t-columns per block (1 lane per column)
#define QCHUNK  32             // q rows staged per TDM chunk
#define NCHUNK  (NQ / QCHUNK)  // 32
// per chunk: QCHUNK*TTILE floats = 16KB moved by ONE tensor_load_to_lds

typedef uint32_t v4u __attribute__((ext_vector_type(4)));
typedef uint32_t v8u __attribute__((ext_vector_type(8)));

__device__ __forceinline__ float fast_rcp(float x) {
    return __builtin_amdgcn_rcpf(x);          // v_rcp_f32, ~1 ulp
}
__device__ __forceinline__ float sigmoidf_(float x) {
    return fast_rcp(1.0f + __expf(-x));
}

// ---- Tensor Data Mover: 2D tile global->LDS (CDNA5 ISA/§8) ----
// D# group0: count=1 | lds_addr | global_addr[56:0] | type=2
// D# group1: data_size=4B; tensor_dim0/1; tile_dim0/1; tensor_dim0_stride
__device__ __forceinline__ void tdm_load_tile_2d(uint32_t lds_addr, uint64_t gaddr) {
    const v4u g0 = {
        1u,                                             // count=1 (valid user D#)
        lds_addr,                                       // LDS byte address
        (uint32_t)(gaddr & 0xFFFFFFFFu),                // global_addr[31:0]
        ((uint32_t)(gaddr >> 32) & 0x01FFFFFFu) | (2u << 30)  // ga[56:32] | type=2
    };
    const v8u g1 = {
        (2u << 16),                 // workgroup_mask=0, data_size=2 (4B), no pad/iter
        ((uint32_t)NT & 0xFFFFu) << 16,   // tensor_dim0[15:0]  (=1024)
        ((uint32_t)NQ & 0xFFFFu) << 16,   // tensor_dim0[31:16]=0 | tensor_dim1[15:0]
        ((uint32_t)TTILE) << 16,          // tensor_dim1[31:16]=0 | tile_dim0=128
        (uint32_t)QCHUNK,                 // tile_dim1=32 | tile_dim2=0
        (uint32_t)NT,                     // tensor_dim0_stride[31:0] = 1024 elements
        0u,                               // dim0_stride[47:32]=0 | dim1_stride[15:0]=0
        0u                                // tensor_dim1_stride[47:16]=0
    };
    asm volatile("tensor_load_to_lds %0, %1" :: "s"(g0), "s"(g1) : "memory");
}

struct ColState {
    float M;    // running max over column (incl. clutter DELTA)
    float S;    // running sum of exp((v - M)*SCALE)
    float Mv;   // running max over valid rows only
    int amax;   // argmax incl. clutter (NQ == clutter)
    int amaxv;  // argmax over valid rows
};

__device__ __forceinline__ void col_process_chunk(const float* __restrict__ sbuf,
                                                  const float* __restrict__ sigs,
                                                  int q0, int tid, ColState& st) {
#pragma unroll
    for (int r = 0; r < QCHUNK; ++r) {
        const int q = q0 + r;
        const float v = sbuf[r * TTILE + tid] * sigs[q];
        if (v > st.M) {
            st.S = st.S * __expf((st.M - v) * SCALE) + 1.0f;
            st.M = v;
            st.amax = q;
        } else {
            st.S += __expf((v - st.M) * SCALE);
        }
        if (v > st.Mv) { st.Mv = v; st.amaxv = q; }
    }
}

__global__ __launch_bounds__(TTILE) void col_pass(const float* __restrict__ assoc,
                                                  const float* __restrict__ dist,
                                                  const float* __restrict__ logits,
                                                  float* __restrict__ out,
                                                  int* __restrict__ amax_ws) {
    __shared__ float buf[2][QCHUNK * TTILE];   // 2 x 16KB double buffer (TDM dest)
    __shared__ float sigs[NQ];                 // 4KB sigmoid table for this m
    __shared__ float redA[TTILE / 32];
    __shared__ float redB[TTILE / 32];

    const int tid = threadIdx.x;
    const int m  = blockIdx.y;
    const int t0 = blockIdx.x * TTILE;
    const int t  = t0 + tid;

    // sigmoid(logits[m, :]) once per block
    for (int i = tid; i < NQ; i += TTILE)
        sigs[i] = sigmoidf_(logits[m * NQ + i]);

    const uint64_t gbase = (uint64_t)(assoc + ((size_t)m * NQ) * NT + t0);
    const uint32_t lds0 = (uint32_t)(uintptr_t)&buf[0][0];
    const uint32_t lds1 = (uint32_t)(uintptr_t)&buf[1][0];

    // wave 0 drives the TDM (EXEC ignored; one engine-side 2D copy per chunk)
    if (tid < 32) {
        tdm_load_tile_2d(lds0, gbase);                                   // chunk 0
        tdm_load_tile_2d(lds1, gbase + (uint64_t)QCHUNK * NT * 4u);      // chunk 1
    }

    ColState st;
    st.M = DELTA; st.S = 1.0f;           // clutter row pre-seeded: exp(0)
    st.Mv = -3.4e38f;
    st.amax = NQ; st.amaxv = 0;

    for (int k = 0; k < NCHUNK; ++k) {
        // in-order TDM completion: <=1 outstanding  ==>  chunk k is done
        if (k < NCHUNK - 1) __builtin_amdgcn_s_wait_tensorcnt(1);
        else                __builtin_amdgcn_s_wait_tensorcnt(0);
        __syncthreads();                                      // visible to all waves
        col_process_chunk(&buf[k & 1][0], sigs, k * QCHUNK, tid, st);
        __syncthreads();                                      // buffer free to reuse
        if (k + 2 < NCHUNK && tid < 32)
            tdm_load_tile_2d((k & 1) ? lds1 : lds0,
                             gbase + (uint64_t)(k + 2) * QCHUNK * NT * 4u);
    }

    // fn = col_softmax clutter row; loc = dist at argmax (if not clutter)
    float fn  = __expf((DELTA - st.M) * SCALE) * fast_rcp(st.S);
    float loc = 0.0f;
    if (st.amax < NQ)
        loc = dist[((size_t)m * NQ + st.amax) * NT + t];

    amax_ws[m * NT + t] = st.amax;
    out[3 + (size_t)NM * NQ * NT + (size_t)m * NT + t] = (float)st.amaxv;

    // block reduce fn / loc -> one atomic each
#pragma unroll
    for (int off = 16; off > 0; off >>= 1) {
        fn  += __shfl_down(fn,  off);
        loc += __shfl_down(loc, off);
    }
    const int wid = tid >> 5, lane = tid & 31;
    if (lane == 0) { redA[wid] = fn; redB[wid] = loc; }
    __syncthreads();
    if (tid == 0) {
        float a = 0.0f, b = 0.0f;
#pragma unroll
        for (int w = 0; w < TTILE / 32; ++w) { a += redA[w]; b += redB[w]; }
        atomicAdd(out + 1, a);   // fn  (MISS_COST/ALPHA == 1)
        atomicAdd(out + 2, b);   // loc
    }
}

// one wave32 per row: online softmax over t, shfl log-sum-exp merge
__global__ __launch_bounds__(256) void row_pass(const float* __restrict__ assoc,
                                                const float* __restrict__ logits,
                                                float* __restrict__ out) {
    __shared__ float red[8];
    const int wid = threadIdx.x >> 5, lane = threadIdx.x & 31;
    const int row = blockIdx.x * 8 + wid;            // 0 .. NM*NQ-1
    const float* rp = assoc + (size_t)row * NT;

    float M = -3.4e38f, S = 0.0f;
    for (int tt = lane; tt < NT; tt += 32) {
        const float v = rp[tt];
        if (v > M) { S = S * __expf((M - v) * SCALE) + 1.0f; M = v; }
        else       { S += __expf((v - M) * SCALE); }
    }
#pragma unroll
    for (int off = 16; off > 0; off >>= 1) {
        const float Mo = __shfl_xor(M, off);
        const float So = __shfl_xor(S, off);
        const float Mn = fmaxf(M, Mo);
        S = S * __expf((M - Mn) * SCALE) + So * __expf((Mo - Mn) * SCALE);
        M = Mn;
    }
    const float Mf = fmaxf(M, DELTA);                 // clutter column
    S = S * __expf((M - Mf) * SCALE) + __expf((DELTA - Mf) * SCALE);
    const float sig = sigmoidf_(logits[row]);
    const float fp = __expf((DELTA - Mf) * SCALE) * fast_rcp(S) * sig;

    if (lane == 0) red[wid] = fp;
    __syncthreads();
    if (threadIdx.x == 0) {
        float a = 0.0f;
#pragma unroll
        for (int w = 0; w < 8; ++w) a += red[w];
        atomicAdd(out + 0, a);   // fp (MISS_COST/ALPHA == 1)
    }
}

// streaming one-hot writer: ha_mask[m,q,t] = (amax[m,t]==q); nontemporal stores
__global__ __launch_bounds__(256) void ha_write(const int* __restrict__ amax_ws,
                                                float* __restrict__ out) {
    const size_t i = (size_t)blockIdx.x * 256 + threadIdx.x;   // < NM*(NQ/4)*NT
    const int t  = (int)(i & (NT - 1));
    const int qg = (int)((i >> 10) & (NQ / 4 - 1));            // group of 4 q's
    const int m  = (int)(i >> 18);
    const int am = amax_ws[m * NT + t];
    float* p = out + 3 + (((size_t)m * NQ + (size_t)qg * 4) * NT) + t;
#pragma unroll
    for (int j = 0; j < 4; ++j)
        __builtin_nontemporal_store((am == qg * 4 + j) ? 1.0f : 0.0f,
                                    p + (size_t)j * NT);
}

__global__ void init_losses(float* __restrict__ out) {
    if (threadIdx.x < 3) out[threadIdx.x] = 0.0f;
}

extern "C" void kernel_launch(void* const* d_in, const int* in_sizes, int n_in,
                              void* d_out, int out_size, void* d_ws, size_t ws_size,
                              hipStream_t stream) {
    (void)in_sizes; (void)n_in; (void)out_size; (void)ws_size;
    const float* assoc  = (const float*)d_in[0];
    const float* dist   = (const float*)d_in[1];
    const float* logits = (const float*)d_in[2];
    float* out = (float*)d_out;
    int* amax_ws = (int*)d_ws;   // NM*NT ints = 256KB

    init_losses<<<1, 32, 0, stream>>>(out);

    dim3 gcol(NT / TTILE, NM);   // (8, 64)
    col_pass<<<gcol, TTILE, 0, stream>>>(assoc, dist, logits, out, amax_ws);

    row_pass<<<(NM * NQ) / 8, 256, 0, stream>>>(assoc, logits, out);

    ha_write<<<(NM * (NQ / 4) * NT) / 256, 256, 0, stream>>>(amax_ws, out);
}